// LocallyConnected2d_11819749999086
// MI455X (gfx1250) — compile-verified
//
#include <hip/hip_runtime.h>

typedef __attribute__((ext_vector_type(2))) float v2f;
typedef __attribute__((ext_vector_type(8))) float v8f;
typedef __attribute__((ext_vector_type(4))) int   v4i;

#define BATCH 16
#define CIN   16
#define HH    64
#define WW    64
#define COUT  16
#define OHH   62
#define OWW   62
#define NPOS  (OHH * OWW)   /* 3844 spatial positions */
#define KTOT  (CIN * 9)     /* 144 = reduction length  */
#define NSTEP (KTOT / 4)    /* 36 wmma f32 16x16x4 steps */

// ---------------------------------------------------------------------------
// Compile-time offset table: for (step s, lane) precompute the 4 element
// offsets this lane needs for its A pair (x input) and B pair (weight),
// with the lane-dependent  mn*stride  terms folded in.  18 KB in .rodata,
// read as one coalesced b128 per lane per step (L2 / WGP$ resident).
//   ck = 4*s + 2*half + t ;  c = ck/9 ; k = ck%9 ; i = k/3 ; j = k%3
//   A offset = mn*65536  + c*4096  + i*64 + j      (relative to x  + y*64+x)
//   B offset = mn*553536 + c*34596 + k             (relative to w  + pos*9 )
// ---------------------------------------------------------------------------
struct alignas(16) OffTab { int v[NSTEP][32][4]; };

constexpr OffTab make_tab() {
    OffTab t{};
    for (int s = 0; s < NSTEP; ++s)
        for (int lane = 0; lane < 32; ++lane) {
            const int half = lane >> 4;
            const int mn   = lane & 15;
            for (int tt = 0; tt < 2; ++tt) {
                const int ck = 4 * s + 2 * half + tt;
                const int c  = ck / 9, k = ck - c * 9;
                const int i  = k / 3,  j = k - i * 3;
                t.v[s][lane][tt]     = mn * (CIN * HH * WW) + c * (HH * WW) + i * WW + j;
                t.v[s][lane][2 + tt] = mn * (CIN * NPOS * 9) + c * (NPOS * 9) + k;
            }
        }
    return t;
}

__device__ constexpr OffTab g_tab = make_tab();

// Global-address-space pointer type: keeps laundered table loads on the
// global path (LOADcnt only) instead of flat (LOADcnt + DScnt + aperture).
typedef const __attribute__((address_space(1))) v4i* gv4i_ptr;

// One wave32 per output position: D[16x16] = A[16x144] * B[144x16]
// via 36x V_WMMA_F32_16X16X4_F32 (full fp32, matches reference precision).
__global__ __launch_bounds__(256)
void lc2d_wmma_kernel(const float* __restrict__ xin,
                      const float* __restrict__ wgt,
                      float* __restrict__ out)
{
    const int lane = threadIdx.x & 31;
    // Force wave id / position into SGPRs: keeps base pointers scalar so the
    // data loads become  global_load_b32 v, voff, s[base] scale_offset  with
    // zero per-step VALU, and the bounds check is a pure scalar branch.
    const int wave = __builtin_amdgcn_readfirstlane(threadIdx.x >> 5);
    const int pos  = blockIdx.x * 8 + wave;
    if (pos >= NPOS) return;              // scalar branch: EXEC stays all-1s

    const int y = pos / OWW;
    const int x = pos - y * OWW;

    const float* __restrict__ xp = xin + y * WW + x;   // scalar base (A)
    const float* __restrict__ wp = wgt + pos * 9;      // scalar base (B)

    // Pin the table base in an SGPR pair ONCE (opaque asm blocks per-step
    // s_get_pc_i64 rematerialization), then cast to address_space(1) so the
    // per-step loads are global_load_b128 with folded immediate offsets.
    const v4i* tg = (const v4i*)&g_tab;
    asm volatile("" : "+s"(tg));
    gv4i_ptr tp = (gv4i_ptr)tg;

    v8f acc = {};

#pragma unroll
    for (int s = 0; s < NSTEP; ++s) {
        const v4i o = tp[s * 32 + lane];   // 1x global b128: s[base]+lane*16, imm s*512
        v2f a, b;
        a.x = xp[o.x];
        a.y = xp[o.y];
        b.x = wp[o.z];
        b.y = wp[o.w];
        // 8-arg form: (neg_a, A, neg_b, B, c_mod, C, reuse_a, reuse_b)
        acc = __builtin_amdgcn_wmma_f32_16x16x4_f32(
            false, a, false, b, (short)0, acc, false, false);
    }

    // D layout (32-bit 16x16): VGPR r -> M = r (lanes 0-15) / r+8 (lanes 16-31),
    // N = lane&15.  out index = (m*COUT + n)*NPOS + pos.
    const int half = lane >> 4;
    const int mn   = lane & 15;
    float* __restrict__ op = out + pos;                     // scalar base
    const int voff = mn * NPOS + half * (8 * COUT * NPOS);  // lane part, once
#pragma unroll
    for (int r = 0; r < 8; ++r)
        op[voff + r * (COUT * NPOS)] = acc[r];              // imm offset r*246016B
}

extern "C" void kernel_launch(void* const* d_in, const int* in_sizes, int n_in,
                              void* d_out, int out_size, void* d_ws, size_t ws_size,
                              hipStream_t stream) {
    const float* xin = (const float*)d_in[0];   // x: [16,16,64,64] f32
    const float* wgt = (const float*)d_in[1];   // weight: [1,16,16,62,62,9] f32
    float* out = (float*)d_out;                 // out: [16,16,62,62] f32

    const int waves_per_block = 8;              // 256 threads = 8 wave32
    const int blocks = (NPOS + waves_per_block - 1) / waves_per_block;  // 481
    lc2d_wmma_kernel<<<blocks, 256, 0, stream>>>(xin, wgt, out);
}